// REModule_12635793784906
// MI455X (gfx1250) — compile-verified
//
#include <hip/hip_runtime.h>

typedef float v2f __attribute__((ext_vector_type(2)));
typedef float v8f __attribute__((ext_vector_type(8)));
typedef int   v4i __attribute__((vector_size(16)));   // GCC-vector int4 (matches builtin proto)
typedef __attribute__((address_space(1))) v4i* g4p;   // global int4*
typedef __attribute__((address_space(3))) v4i* l4p;   // LDS int4*

#define B_   16
#define N_   64
#define IN_  768
#define H_   256
#define O_   16
#define P_   (N_ * (N_ - 1))   // 4032 pairs per batch
#define MTOT (B_ * N_)         // 1024 tokens

// fp32 WMMA: D(16x16,f32) = A(16x4,f32) * B(4x16,f32) + C
static __device__ __forceinline__ v8f wmma4(v2f a, v2f b, v8f c) {
  return __builtin_amdgcn_wmma_f32_16x16x4_f32(false, a, false, b, (short)0, c,
                                               false, false);
}

// Async 16-byte copy global -> LDS (tracked on ASYNCcnt, no VGPR data path)
static __device__ __forceinline__ void async_copy16(float* lds_dst,
                                                    const float* gsrc) {
#if __has_builtin(__builtin_amdgcn_global_load_async_to_lds_b128)
  v4i* gsrc_nc = (v4i*)gsrc;  // drop const / re-pointee in generic AS
  __builtin_amdgcn_global_load_async_to_lds_b128((g4p)gsrc_nc, (l4p)lds_dst, 0,
                                                 0);
#else
  unsigned loff = (unsigned)(size_t)(l4p)lds_dst;
  asm volatile("global_load_async_to_lds_b128 %0, %1, off" ::"v"(loff),
               "v"(gsrc)
               : "memory");
#endif
}

static __device__ __forceinline__ void wait_async0() {
#if __has_builtin(__builtin_amdgcn_s_wait_asynccnt)
  __builtin_amdgcn_s_wait_asynccnt(0);
#else
  asm volatile("s_wait_asynccnt 0x0" ::: "memory");
#endif
}

// --------------------------------------------------------------------------
// Fused head/tail GEMM:  D = relu?(A @ W^T + bias)
// A: MTOT x K (row-major), W: H_ x K (row-major, so B[k][n] = W[n][k])
// One wave per 16x16 output tile; 2*64*16 = 2048 wave-tiles total.
// --------------------------------------------------------------------------
__global__ __launch_bounds__(256) void mlp_gemm(
    const float* __restrict__ Ah, const float* __restrict__ At,
    const float* __restrict__ Wh, const float* __restrict__ bh,
    const float* __restrict__ Wt, const float* __restrict__ bt,
    float* __restrict__ Dh, float* __restrict__ Dt, int K, int do_relu) {
  const int NT = H_ / 16;                       // 16 N-tiles
  const int tiles_per_mat = (MTOT / 16) * NT;   // 1024
  int wv   = blockIdx.x * (blockDim.x >> 5) + (threadIdx.x >> 5);
  int lane = threadIdx.x & 31;

  int sel = (wv >= tiles_per_mat) ? 1 : 0;
  int t   = sel ? (wv - tiles_per_mat) : wv;
  int mt = t / NT, nt = t % NT;

  const float* A    = sel ? At : Ah;
  const float* W    = sel ? Wt : Wh;
  const float* bias = sel ? bt : bh;
  float*       D    = sel ? Dt : Dh;

  int m0 = mt * 16, n0 = nt * 16;
  int lrow = lane & 15;
  int kq   = (lane >> 4) << 1;   // 0 for lanes 0-15, 2 for lanes 16-31

  const float* arow = A + (size_t)(m0 + lrow) * K + kq;
  const float* brow = W + (size_t)(n0 + lrow) * K + kq;

  v8f acc = {};
#pragma unroll 4
  for (int k = 0; k < K; k += 4) {
    v2f a, b;
    a.x = arow[k];  a.y = arow[k + 1];
    b.x = brow[k];  b.y = brow[k + 1];
    acc = wmma4(a, b, acc);
  }

  int   cn  = n0 + lrow;
  int   cmb = m0 + ((lane >> 4) << 3);
  float bv  = bias[cn];
#pragma unroll
  for (int r = 0; r < 8; ++r) {
    float v = acc[r] + bv;
    if (do_relu) v = fmaxf(v, 0.0f);
    D[(size_t)(cmb + r) * H_ + cn] = v;
  }
}

// --------------------------------------------------------------------------
// lh = head @ W_lin[:, :H]^T ; lt = tail @ W_lin[:, H:]^T   (N = O_ = 16)
// --------------------------------------------------------------------------
__global__ __launch_bounds__(256) void lin_gemm(
    const float* __restrict__ Head, const float* __restrict__ Tail,
    const float* __restrict__ Wlin, float* __restrict__ Lh,
    float* __restrict__ Lt) {
  int wv   = blockIdx.x * (blockDim.x >> 5) + (threadIdx.x >> 5);
  int lane = threadIdx.x & 31;
  int sel  = (wv >= MTOT / 16) ? 1 : 0;
  int mt   = sel ? (wv - MTOT / 16) : wv;

  const float* A = sel ? Tail : Head;
  const float* W = Wlin + (sel ? H_ : 0);   // row stride 2*H_
  float*       D = sel ? Lt : Lh;

  int m0   = mt * 16;
  int lrow = lane & 15;
  int kq   = (lane >> 4) << 1;

  const float* arow = A + (size_t)(m0 + lrow) * H_ + kq;
  const float* brow = W + (size_t)lrow * (2 * H_) + kq;  // B[k][n] = Wlin[n][k]

  v8f acc = {};
#pragma unroll 4
  for (int k = 0; k < H_; k += 4) {
    v2f a, b;
    a.x = arow[k];  a.y = arow[k + 1];
    b.x = brow[k];  b.y = brow[k + 1];
    acc = wmma4(a, b, acc);
  }

  int cn  = lrow;
  int cmb = m0 + ((lane >> 4) << 3);
#pragma unroll
  for (int r = 0; r < 8; ++r)
    D[(size_t)(cmb + r) * O_ + cn] = acc[r];
}

// --------------------------------------------------------------------------
// One workgroup per (b, o):
//   (async DMA tail_b -> LDS, overlapped with phase 1)
//   M  = head_b(64x256) @ W_bil[o](256x256)       -> LDS
//   S  = M(64x256)      @ tail_b^T(256x64)        -> LDS
//   scores[b, p(i,j), o] = S[i][j] + b_bil[o] + lh[b,i,o] + lt[b,j,o]
// --------------------------------------------------------------------------
__global__ __launch_bounds__(256) void pair_kernel(
    const float* __restrict__ Head, const float* __restrict__ Tail,
    const float* __restrict__ Wbil, const float* __restrict__ bbil,
    const float* __restrict__ Lh, const float* __restrict__ Lt,
    float* __restrict__ scores) {
  __shared__ float lds_tail[N_ * H_];  // 64 KB
  __shared__ float lds_M[N_ * H_];     // 64 KB
  __shared__ float lds_S[N_ * N_];     // 16 KB

  int b = blockIdx.x >> 4;
  int o = blockIdx.x & 15;

  int tid  = threadIdx.x;
  int wv   = tid >> 5;
  int lane = tid & 31;

  const float* headb = Head + (size_t)b * N_ * H_;
  const float* tailb = Tail + (size_t)b * N_ * H_;
  const float* Wb    = Wbil + (size_t)o * H_ * H_;

  // Kick off async DMA of tail_b into LDS; drains while phase 1 computes.
#pragma unroll
  for (int idx = tid; idx < N_ * H_ / 4; idx += 256)
    async_copy16(lds_tail + idx * 4, tailb + idx * 4);

  int lrow = lane & 15;
  int kq   = (lane >> 4) << 1;

  // Phase 1: M = head_b @ Wb   (64 tiles, 8 per wave) — no lds_tail use here
  for (int t = wv; t < (N_ / 16) * (H_ / 16); t += 8) {
    int mt = t >> 4, nt = t & 15;
    int m0 = mt * 16, n0 = nt * 16;
    const float* arow = headb + (size_t)(m0 + lrow) * H_ + kq;
    const float* bcol = Wb + n0 + lrow;   // B[k][n] = Wb[k*H_ + n]
    v8f acc = {};
#pragma unroll 4
    for (int k = 0; k < H_; k += 4) {
      v2f a, bb;
      a.x  = arow[k];               a.y  = arow[k + 1];
      bb.x = bcol[(size_t)(k + kq) * H_];
      bb.y = bcol[(size_t)(k + kq + 1) * H_];
      acc = wmma4(a, bb, acc);
    }
    int cn  = n0 + lrow;
    int cmb = m0 + ((lane >> 4) << 3);
#pragma unroll
    for (int r = 0; r < 8; ++r)
      lds_M[(cmb + r) * H_ + cn] = acc[r];
  }

  // Each wave waits for its own async DMA, then workgroup barrier.
  wait_async0();
  __syncthreads();

  // Phase 2: S = M @ tail^T   (16 tiles, 2 per wave)
  for (int t = wv; t < (N_ / 16) * (N_ / 16); t += 8) {
    int mt = t >> 2, nt = t & 3;
    int m0 = mt * 16, n0 = nt * 16;
    const float* arow = lds_M    + (m0 + lrow) * H_ + kq;
    const float* brow = lds_tail + (n0 + lrow) * H_ + kq;  // B[k][n]=tail[n][k]
    v8f acc = {};
#pragma unroll 4
    for (int k = 0; k < H_; k += 4) {
      v2f a, bb;
      a.x  = arow[k];  a.y  = arow[k + 1];
      bb.x = brow[k];  bb.y = brow[k + 1];
      acc = wmma4(a, bb, acc);
    }
    int cn  = n0 + lrow;
    int cmb = m0 + ((lane >> 4) << 3);
#pragma unroll
    for (int r = 0; r < 8; ++r)
      lds_S[(cmb + r) * N_ + cn] = acc[r];
  }
  __syncthreads();

  // Phase 3: epilogue + pair scatter (p = 63*i + j - (j>i))
  float bbo = bbil[o];
  for (int g = tid; g < N_ * N_; g += 256) {
    int i = g >> 6, j = g & 63;
    if (i == j) continue;
    int p = i * 63 + j - (j > i ? 1 : 0);
    float v = lds_S[g] + bbo + Lh[(size_t)(b * N_ + i) * O_ + o] +
              Lt[(size_t)(b * N_ + j) * O_ + o];
    __builtin_nontemporal_store(v, &scores[((size_t)b * P_ + p) * O_ + o]);
  }
}

// --------------------------------------------------------------------------
// pos[b,p] = (positions[b, i], positions[b, j])
// --------------------------------------------------------------------------
__global__ void pos_kernel(const int* __restrict__ positions,
                           int* __restrict__ pos_out) {
  int t = blockIdx.x * blockDim.x + threadIdx.x;
  if (t >= B_ * P_) return;
  int b = t / P_, p = t % P_;
  int i = p / 63, r = p % 63;
  int j = r + (r >= i ? 1 : 0);
  unsigned pi = (unsigned)positions[b * N_ + i];
  unsigned pj = (unsigned)positions[b * N_ + j];
  // little-endian pack: pi -> pos_out[2t], pj -> pos_out[2t+1]
  unsigned long long pr = ((unsigned long long)pj << 32) | (unsigned long long)pi;
  __builtin_nontemporal_store(pr, (unsigned long long*)pos_out + t);
}

// --------------------------------------------------------------------------
extern "C" void kernel_launch(void* const* d_in, const int* in_sizes, int n_in,
                              void* d_out, int out_size, void* d_ws,
                              size_t ws_size, hipStream_t stream) {
  (void)in_sizes; (void)n_in; (void)out_size; (void)ws_size;

  const float* x     = (const float*)d_in[0];
  const int*   posi  = (const int*)d_in[1];
  const float* w_h1  = (const float*)d_in[2];
  const float* b_h1  = (const float*)d_in[3];
  const float* w_h2  = (const float*)d_in[4];
  const float* b_h2  = (const float*)d_in[5];
  const float* w_t1  = (const float*)d_in[6];
  const float* b_t1  = (const float*)d_in[7];
  const float* w_t2  = (const float*)d_in[8];
  const float* b_t2  = (const float*)d_in[9];
  const float* W_bil = (const float*)d_in[10];
  const float* b_bil = (const float*)d_in[11];
  const float* W_lin = (const float*)d_in[12];

  float* ws      = (float*)d_ws;
  float* h1_head = ws;  ws += (size_t)MTOT * H_;
  float* h1_tail = ws;  ws += (size_t)MTOT * H_;
  float* head    = ws;  ws += (size_t)MTOT * H_;
  float* tail    = ws;  ws += (size_t)MTOT * H_;
  float* lh      = ws;  ws += (size_t)MTOT * O_;
  float* lt      = ws;  ws += (size_t)MTOT * O_;

  int*   pos_out = (int*)d_out;
  float* scores  = (float*)d_out + (size_t)B_ * P_ * 2;

  // layer 1 (relu) for head+tail: 2048 wave-tiles, 8 waves/block
  mlp_gemm<<<256, 256, 0, stream>>>(x, x, w_h1, b_h1, w_t1, b_t1, h1_head,
                                    h1_tail, IN_, 1);
  // layer 2
  mlp_gemm<<<256, 256, 0, stream>>>(h1_head, h1_tail, w_h2, b_h2, w_t2, b_t2,
                                    head, tail, H_, 0);
  // linear terms lh / lt: 128 wave-tiles
  lin_gemm<<<16, 256, 0, stream>>>(head, tail, W_lin, lh, lt);
  // bilinear pair scores: one workgroup per (b, o)
  pair_kernel<<<B_ * O_, 256, 0, stream>>>(head, tail, W_bil, b_bil, lh, lt,
                                           scores);
  // pair positions
  pos_kernel<<<(B_ * P_ + 255) / 256, 256, 0, stream>>>(posi, pos_out);
}